// PhraseClassifier_43576738185614
// MI455X (gfx1250) — compile-verified
//
#include <hip/hip_runtime.h>
#include <math.h>

// MI455X / gfx1250 (wave32). Gather-bound problem: 400MB of L2-resident random
// row gathers, 0.4 GFLOP. Design: wave-per-phrase coalesced float4 gathers,
// W staged in LDS, prefetch of next phrase, WMMA-based wave reduction.

typedef float v2f __attribute__((ext_vector_type(2)));
typedef float v8f __attribute__((ext_vector_type(8)));

#define NBLOCKS 512
#define NTHREADS 256
#define WAVES_PER_BLOCK (NTHREADS / 32)

__device__ __forceinline__ float dot4(float4 a, float4 b) {
  return a.x * b.x + a.y * b.y + a.z * b.z + a.w * b.w;
}

__global__ __launch_bounds__(NTHREADS) void phrase_loss_partial_kernel(
    const float* __restrict__ fwd, const float* __restrict__ bwd,
    const int* __restrict__ bids, const int* __restrict__ begins,
    const int* __restrict__ ends, const float* __restrict__ targets,
    const float* __restrict__ W, const float* __restrict__ b,
    float* __restrict__ partial, int N) {
  constexpr int H = 512;   // hidden size (per reference)
  constexpr int Bdim = 32; // batch size (per reference)

  __shared__ __align__(16) float sW[4 * H]; // 8 KB: full weight vector
  __shared__ float swave[WAVES_PER_BLOCK];

  const int tid = threadIdx.x;
  const int lane = tid & 31;
  const int wave = tid >> 5;

  // Stage W into LDS (coalesced, 8 floats per thread).
#pragma unroll
  for (int i = 0; i < (4 * H) / NTHREADS; ++i)
    sW[tid + i * NTHREADS] = W[tid + i * NTHREADS];
  __syncthreads();

  const float bb = b[0];
  const float4* w4 = (const float4*)sW;

  float lsum = 0.0f;
  const int stride = gridDim.x * WAVES_PER_BLOCK;

  for (int p = (int)blockIdx.x * WAVES_PER_BLOCK + wave; p < N; p += stride) {
    // ---- prefetch next phrase's 4 gathered rows (global_prefetch_b8) ----
    const int pn = p + stride;
    if (pn < N) {
      const int nb = bids[pn], nbg = begins[pn], nen = ends[pn];
      const float* a0 = fwd + ((size_t)(nbg - 1) * Bdim + nb) * H;
      const float* a1 = fwd + ((size_t)(nen - 1) * Bdim + nb) * H;
      const float* a2 = bwd + ((size_t)nen * Bdim + nb) * H;
      const float* a3 = bwd + ((size_t)nbg * Bdim + nb) * H;
      __builtin_prefetch(a0 + lane * 16, 0, 0);
      __builtin_prefetch(a1 + lane * 16, 0, 0);
      __builtin_prefetch(a2 + lane * 16, 0, 0);
      __builtin_prefetch(a3 + lane * 16, 0, 0);
    }

    // ---- gather + dot (coalesced float4 loads, LDS W, conflict-free) ----
    const int bidv = bids[p], bg = begins[p], en = ends[p];
    const float tgt = targets[p];
    const float4* r0 = (const float4*)(fwd + ((size_t)(bg - 1) * Bdim + bidv) * H);
    const float4* r1 = (const float4*)(fwd + ((size_t)(en - 1) * Bdim + bidv) * H);
    const float4* r2 = (const float4*)(bwd + ((size_t)en * Bdim + bidv) * H);
    const float4* r3 = (const float4*)(bwd + ((size_t)bg * Bdim + bidv) * H);

    float acc = 0.0f;
#pragma unroll
    for (int k = 0; k < 4; ++k) {
      const int idx = lane + k * 32; // 0..127 float4s per 512-float row
      acc += dot4(r0[idx], w4[idx + 0 * 128]);
      acc += dot4(r1[idx], w4[idx + 1 * 128]);
      acc += dot4(r2[idx], w4[idx + 2 * 128]);
      acc += dot4(r3[idx], w4[idx + 3 * 128]);
    }

    // ---- wave-wide reduction of 32 lane partials ----
    float tot;
#if defined(__gfx1250__) && __has_builtin(__builtin_amdgcn_wmma_f32_16x16x4_f32)
    {
      // A (16x4 f32, 2 VGPRs): vgpr0 = lane partial, vgpr1 = 0.
      // B (4x16 f32): all ones -> layout-independent:
      //   D[m][n] = sum_k A[m][k] = p_m + p_{m+16}.
      v2f a;
      a[0] = acc;
      a[1] = 0.0f;
      v2f bm;
      bm[0] = 1.0f;
      bm[1] = 1.0f;
      v8f c = {};
      c = __builtin_amdgcn_wmma_f32_16x16x4_f32(
          /*neg_a=*/false, a, /*neg_b=*/false, bm,
          /*c_mod=*/(short)0, c, /*reuse_a=*/false, /*reuse_b=*/false);
      // Per lane: c[r] = D[row][laneN]; lanes 0-15 hold rows 0-7,
      // lanes 16-31 hold rows 8-15 -> sum 8 regs then exchange halves.
      float s = c[0] + c[1] + c[2] + c[3] + c[4] + c[5] + c[6] + c[7];
      s += __shfl_xor(s, 16, 32);
      tot = s;
    }
#else
    {
      float s = acc;
      for (int off = 16; off > 0; off >>= 1) s += __shfl_xor(s, off, 32);
      tot = s;
    }
#endif

    // ---- BCE-with-logits (every lane holds same total; redundant ok) ----
    const float logit = tot + bb;
    const float l =
        fmaxf(logit, 0.0f) - logit * tgt + log1pf(expf(-fabsf(logit)));
    lsum += l;
  }

  if (lane == 0) swave[wave] = lsum;
  __syncthreads();
  if (tid == 0) {
    float s = 0.0f;
#pragma unroll
    for (int i = 0; i < WAVES_PER_BLOCK; ++i) s += swave[i];
    partial[blockIdx.x] = s;
  }
}

__global__ __launch_bounds__(256) void reduce_partials_kernel(
    const float* __restrict__ partial, float* __restrict__ out, int nparts,
    float invN) {
  __shared__ float red[8];
  const int tid = threadIdx.x;
  const int lane = tid & 31;
  const int wave = tid >> 5;
  float s = 0.0f;
  for (int i = tid; i < nparts; i += 256) s += partial[i];
  for (int off = 16; off > 0; off >>= 1) s += __shfl_xor(s, off, 32);
  if (lane == 0) red[wave] = s;
  __syncthreads();
  if (tid == 0) {
    float t = 0.0f;
#pragma unroll
    for (int i = 0; i < 8; ++i) t += red[i];
    out[0] = t * invN;
  }
}

extern "C" void kernel_launch(void* const* d_in, const int* in_sizes, int n_in,
                              void* d_out, int out_size, void* d_ws,
                              size_t ws_size, hipStream_t stream) {
  const float* fwd = (const float*)d_in[0];
  const float* bwd = (const float*)d_in[1];
  const int* bids = (const int*)d_in[2];
  const int* begins = (const int*)d_in[3];
  const int* ends = (const int*)d_in[4];
  const float* targets = (const float*)d_in[5];
  const float* W = (const float*)d_in[6];
  const float* b = (const float*)d_in[7];
  const int N = in_sizes[2];

  float* partial = (float*)d_ws; // NBLOCKS floats, fully written each call

  phrase_loss_partial_kernel<<<NBLOCKS, NTHREADS, 0, stream>>>(
      fwd, bwd, bids, begins, ends, targets, W, b, partial, N);
  reduce_partials_kernel<<<1, 256, 0, stream>>>(partial, (float*)d_out, NBLOCKS,
                                                1.0f / (float)N);
}